// ScoreBasedDiffusion_18528488915130
// MI455X (gfx1250) — compile-verified
//
#include <hip/hip_runtime.h>
#include <math.h>

// ScoreBasedDiffusion: two MLPs (D=32, H=256, B=2048) + exact Jacobian trace /
// divergence via the identity  trace = rowdot(d1 @ (W2 ⊙ (W3@W1)^T), d2).
// All GEMMs use V_WMMA_F32_16X16X4_F32 (native fp32 matrix op on CDNA5).
//
// Workspace layout (floats):
//   [0 .. 65536)                      A_s  = sW2 ⊙ (sW3@sW1)^T          (256x256)
//   [65536 .. 2162688)                Z_i  = cW2 ⊙ (cW3_i@cW1)^T, i<32  (32 x 256x256)
//   [2162688 .. 2686976)              h1c  activations [2048,256]
//   [2686976 .. 3211264)              h2c  activations [2048,256]
// Total: 12,845,056 bytes.

typedef float v2f __attribute__((ext_vector_type(2)));
typedef float v8f __attribute__((ext_vector_type(8)));

#define D_IN   32
#define H_DIM  256
#define B_SZ   2048

// ---------------------------------------------------------------------------
// One 16x16 output tile of  X[16,K] @ W[K,N] (cols n0..n0+15), A from LDS
// stored transposed as Xt[k*16 + row].  If `deriv`, A elements become 1-a*a
// (tanh'), letting us keep only activations in LDS.
// A layout (f32 16x16x4): lane m=l&15, half=l>>4: v0=A[m,k0+2h], v1=A[m,k0+2h+1]
// B layout: v0=B[k0+2h, n0+m], v1=B[k0+2h+1, n0+m]
// ---------------------------------------------------------------------------
__device__ __forceinline__ v8f gemm_tile_T(const float* __restrict__ Xt,
                                           const float* __restrict__ W,
                                           int N, int n0, int K,
                                           int m, int half, bool deriv) {
  v8f acc = {};
  for (int k0 = 0; k0 < K; k0 += 4) {
    v2f a, b;
    a.x = Xt[(k0 + 2 * half) * 16 + m];
    a.y = Xt[(k0 + 2 * half + 1) * 16 + m];
    if (deriv) { a.x = 1.0f - a.x * a.x; a.y = 1.0f - a.y * a.y; }
    b.x = W[(k0 + 2 * half) * N + n0 + m];
    b.y = W[(k0 + 2 * half + 1) * N + n0 + m];
    acc = __builtin_amdgcn_wmma_f32_16x16x4_f32(false, a, false, b,
                                                (short)0, acc, false, false);
  }
  return acc;
}

// First layer: A rows come straight from global x[(b0+m)*32 + k] (K=32).
__device__ __forceinline__ v8f gemm_tile_x(const float* __restrict__ x, int b0,
                                           const float* __restrict__ W,
                                           int N, int n0, int m, int half) {
  v8f acc = {};
  for (int k0 = 0; k0 < D_IN; k0 += 4) {
    v2f a = *(const v2f*)(x + (b0 + m) * D_IN + k0 + 2 * half);
    v2f b;
    b.x = W[(k0 + 2 * half) * N + n0 + m];
    b.y = W[(k0 + 2 * half + 1) * N + n0 + m];
    acc = __builtin_amdgcn_wmma_f32_16x16x4_f32(false, a, false, b,
                                                (short)0, acc, false, false);
  }
  return acc;
}

// ---------------------------------------------------------------------------
// Precompute A_s and the 32 Z_i matrices (69 M MACs total - VALU is fine).
// block 0 -> A_s ; block 1+i -> Z_i
// ---------------------------------------------------------------------------
__global__ __launch_bounds__(256) void precompute_kernel(
    const float* __restrict__ sW1, const float* __restrict__ sW2,
    const float* __restrict__ sW3, const float* __restrict__ cW1,
    const float* __restrict__ cW2, const float* __restrict__ cW3,
    float* __restrict__ ws) {
  const int blk = blockIdx.x;
  if (blk == 0) {
    for (int idx = threadIdx.x; idx < H_DIM * H_DIM; idx += blockDim.x) {
      const int p = idx >> 8, q = idx & 255;
      float s = 0.0f;
      for (int k = 0; k < D_IN; ++k)
        s += sW3[q * D_IN + k] * sW1[k * H_DIM + p];       // M[q,p]
      ws[idx] = sW2[p * H_DIM + q] * s;                    // A_s[p,q]
    }
  } else {
    const int i = blk - 1;
    float* Z = ws + 65536 + i * 65536;
    for (int idx = threadIdx.x; idx < H_DIM * H_DIM; idx += blockDim.x) {
      const int p = idx >> 8, q = idx & 255;
      float s = 0.0f;
      for (int j = 0; j < D_IN; ++j)
        s += cW3[q * 1024 + i * D_IN + j] * cW1[j * H_DIM + p];  // Mc_i[q,p]
      Z[idx] = cW2[p * H_DIM + q] * s;                           // Z_i[p,q]
    }
  }
}

// ---------------------------------------------------------------------------
// Score net: forward + exact Jacobian trace.  One wave per 16-row batch tile.
// ---------------------------------------------------------------------------
__global__ __launch_bounds__(32) void score_kernel(
    const float* __restrict__ x,
    const float* __restrict__ sW1, const float* __restrict__ sb1,
    const float* __restrict__ sW2, const float* __restrict__ sb2,
    const float* __restrict__ sW3, const float* __restrict__ sb3,
    const float* __restrict__ As,
    float* __restrict__ out_score, float* __restrict__ out_trace) {
  __shared__ float h1t[H_DIM * 16];   // transposed: [col][row]
  __shared__ float h2t[H_DIM * 16];
  const int lane = threadIdx.x;
  const int m = lane & 15, half = lane >> 4;
  const int b0 = blockIdx.x * 16;

  // layer 1: h1 = tanh(x @ sW1 + sb1)
  for (int n0 = 0; n0 < H_DIM; n0 += 16) {
    v8f acc = gemm_tile_x(x, b0, sW1, H_DIM, n0, m, half);
    const float bias = sb1[n0 + m];
    for (int r = 0; r < 8; ++r)
      h1t[(n0 + m) * 16 + r + 8 * half] = tanhf(acc[r] + bias);
  }
  __syncthreads();
  // layer 2: h2 = tanh(h1 @ sW2 + sb2)
  for (int n0 = 0; n0 < H_DIM; n0 += 16) {
    v8f acc = gemm_tile_T(h1t, sW2, H_DIM, n0, H_DIM, m, half, false);
    const float bias = sb2[n0 + m];
    for (int r = 0; r < 8; ++r)
      h2t[(n0 + m) * 16 + r + 8 * half] = tanhf(acc[r] + bias);
  }
  __syncthreads();
  // head: score = h2 @ sW3 + sb3
  for (int n0 = 0; n0 < D_IN; n0 += 16) {
    v8f acc = gemm_tile_T(h2t, sW3, D_IN, n0, H_DIM, m, half, false);
    const float bias = sb3[n0 + m];
    for (int r = 0; r < 8; ++r)
      out_score[(b0 + r + 8 * half) * D_IN + n0 + m] = acc[r] + bias;
  }
  // trace: U = d1 @ A_s ; trace[b] = rowdot(U, d2), d = 1 - h^2 on the fly
  float tr[8];
  for (int r = 0; r < 8; ++r) tr[r] = 0.0f;
  for (int n0 = 0; n0 < H_DIM; n0 += 16) {
    v8f acc = gemm_tile_T(h1t, As, H_DIM, n0, H_DIM, m, half, true);
    for (int r = 0; r < 8; ++r) {
      const float h2v = h2t[(n0 + m) * 16 + r + 8 * half];
      tr[r] += acc[r] * (1.0f - h2v * h2v);
    }
  }
  for (int r = 0; r < 8; ++r)
    for (int off = 8; off >= 1; off >>= 1)
      tr[r] += __shfl_xor(tr[r], off, 32);
  if (m == 0)
    for (int r = 0; r < 8; ++r)
      out_trace[b0 + r + 8 * half] = tr[r];
}

// ---------------------------------------------------------------------------
// Covariance net forward: cov [B,1024]; dumps h1c,h2c to workspace.
// ---------------------------------------------------------------------------
__global__ __launch_bounds__(32) void cov_fwd_kernel(
    const float* __restrict__ x,
    const float* __restrict__ cW1, const float* __restrict__ cb1,
    const float* __restrict__ cW2, const float* __restrict__ cb2,
    const float* __restrict__ cW3, const float* __restrict__ cb3,
    float* __restrict__ out_cov,
    float* __restrict__ wh1, float* __restrict__ wh2) {
  __shared__ float h1t[H_DIM * 16];
  __shared__ float h2t[H_DIM * 16];
  const int lane = threadIdx.x;
  const int m = lane & 15, half = lane >> 4;
  const int b0 = blockIdx.x * 16;

  for (int n0 = 0; n0 < H_DIM; n0 += 16) {
    v8f acc = gemm_tile_x(x, b0, cW1, H_DIM, n0, m, half);
    const float bias = cb1[n0 + m];
    for (int r = 0; r < 8; ++r)
      h1t[(n0 + m) * 16 + r + 8 * half] = tanhf(acc[r] + bias);
  }
  __syncthreads();
  for (int n0 = 0; n0 < H_DIM; n0 += 16) {
    v8f acc = gemm_tile_T(h1t, cW2, H_DIM, n0, H_DIM, m, half, false);
    const float bias = cb2[n0 + m];
    for (int r = 0; r < 8; ++r)
      h2t[(n0 + m) * 16 + r + 8 * half] = tanhf(acc[r] + bias);
  }
  __syncthreads();
  // head: cov = h2 @ cW3 + cb3   ([16,256]@[256,1024])
  for (int n0 = 0; n0 < 1024; n0 += 16) {
    v8f acc = gemm_tile_T(h2t, cW3, 1024, n0, H_DIM, m, half, false);
    const float bias = cb3[n0 + m];
    for (int r = 0; r < 8; ++r)
      out_cov[(b0 + r + 8 * half) * 1024 + n0 + m] = acc[r] + bias;
  }
  // dump activations for the divergence pass
  for (int idx = lane; idx < 16 * H_DIM; idx += 32) {
    const int row = idx >> 8, col = idx & 255;
    wh1[(b0 + row) * H_DIM + col] = h1t[col * 16 + row];
    wh2[(b0 + row) * H_DIM + col] = h2t[col * 16 + row];
  }
}

// ---------------------------------------------------------------------------
// cov_div[b,i] = rowdot(d1c @ Z_i, d2c).  grid = (128 batch tiles, 32 i).
// This is the dominant compute: 32 x [2048,256]@[256,256] fp32 WMMA GEMMs.
// ---------------------------------------------------------------------------
__global__ __launch_bounds__(32) void cov_div_kernel(
    const float* __restrict__ wh1, const float* __restrict__ wh2,
    const float* __restrict__ Z, float* __restrict__ out_covdiv) {
  __shared__ float h1t[H_DIM * 16];
  __shared__ float h2t[H_DIM * 16];
  const int lane = threadIdx.x;
  const int m = lane & 15, half = lane >> 4;
  const int b0 = blockIdx.x * 16;
  const int i = blockIdx.y;
  const float* Zi = Z + i * 65536;

  for (int idx = lane; idx < 16 * H_DIM; idx += 32) {
    const int row = idx >> 8, col = idx & 255;
    h1t[col * 16 + row] = wh1[(b0 + row) * H_DIM + col];
    h2t[col * 16 + row] = wh2[(b0 + row) * H_DIM + col];
  }
  __syncthreads();

  float cd[8];
  for (int r = 0; r < 8; ++r) cd[r] = 0.0f;
  for (int n0 = 0; n0 < H_DIM; n0 += 16) {
    v8f acc = gemm_tile_T(h1t, Zi, H_DIM, n0, H_DIM, m, half, true);
    for (int r = 0; r < 8; ++r) {
      const float h2v = h2t[(n0 + m) * 16 + r + 8 * half];
      cd[r] += acc[r] * (1.0f - h2v * h2v);
    }
  }
  for (int r = 0; r < 8; ++r)
    for (int off = 8; off >= 1; off >>= 1)
      cd[r] += __shfl_xor(cd[r], off, 32);
  if (m == 0)
    for (int r = 0; r < 8; ++r)
      out_covdiv[(b0 + r + 8 * half) * D_IN + i] = cd[r];
}

// ---------------------------------------------------------------------------
extern "C" void kernel_launch(void* const* d_in, const int* in_sizes, int n_in,
                              void* d_out, int out_size, void* d_ws, size_t ws_size,
                              hipStream_t stream) {
  const float* x   = (const float*)d_in[0];
  const float* sW1 = (const float*)d_in[1];
  const float* sb1 = (const float*)d_in[2];
  const float* sW2 = (const float*)d_in[3];
  const float* sb2 = (const float*)d_in[4];
  const float* sW3 = (const float*)d_in[5];
  const float* sb3 = (const float*)d_in[6];
  const float* cW1 = (const float*)d_in[7];
  const float* cb1 = (const float*)d_in[8];
  const float* cW2 = (const float*)d_in[9];
  const float* cb2 = (const float*)d_in[10];
  const float* cW3 = (const float*)d_in[11];
  const float* cb3 = (const float*)d_in[12];

  float* ws  = (float*)d_ws;
  float* As  = ws;                  // 65536
  float* Z   = ws + 65536;          // 32 * 65536
  float* wh1 = ws + 2162688;        // 2048*256
  float* wh2 = ws + 2686976;        // 2048*256

  float* out        = (float*)d_out;
  float* o_score    = out;              // [2048,32]
  float* o_trace    = out + 65536;      // [2048]
  float* o_cov      = out + 67584;      // [2048,1024]
  float* o_covdiv   = out + 2164736;    // [2048,32]

  precompute_kernel<<<33, 256, 0, stream>>>(sW1, sW2, sW3, cW1, cW2, cW3, ws);
  score_kernel<<<B_SZ / 16, 32, 0, stream>>>(x, sW1, sb1, sW2, sb2, sW3, sb3,
                                             As, o_score, o_trace);
  cov_fwd_kernel<<<B_SZ / 16, 32, 0, stream>>>(x, cW1, cb1, cW2, cb2, cW3, cb3,
                                               o_cov, wh1, wh2);
  cov_div_kernel<<<dim3(B_SZ / 16, 32), 32, 0, stream>>>(wh1, wh2, Z, o_covdiv);
}